// DOnePoleCell_83468394431111
// MI455X (gfx1250) — compile-verified
//
#include <hip/hip_runtime.h>

// -----------------------------------------------------------------------------
// One-pole IIR scan  out_t = b0*x_t + s_t ;  s_{t+1} = b1*x_t + a1c*out_t
// Equivalent linear recurrence: s_{t+1} = a1c*s_t + c*x_t,  c = b1 + a1c*b0.
//
// Strategy (MI455X, gfx1250, wave32):
//  - memory-bound kernel (~0.6 flop/byte): target is streaming 768 MiB ~ 33us
//  - 16-step blocks become an affine map -> 16x16x16 f32 matmul per
//    (16 time x 16 channel) tile, executed as 4 chained V_WMMA_F32_16X16X4_F32
//  - carry enters through the WMMA C accumulator (C[r,ch] = a1^row * s0[ch])
//  - cross-segment parallelism via 3-pass linear scan decomposition
// -----------------------------------------------------------------------------

typedef __attribute__((ext_vector_type(2))) float v2f;
typedef __attribute__((ext_vector_type(8))) float v8f;

#define NSEG 32          // segments along T (parallel chains per channel)
#define WAVES_PER_BLOCK 8

__device__ __forceinline__ float clamp1(float a) {
    return fminf(fmaxf(a, -1.0f), 1.0f);
}

// ---------------------------------------------------------------------------
// Pass 1: per (channel, segment): run recurrence with s0 = 0, emit final s.
// carry[seg*B + ch] = sum_k c * a1^(L-1-k) * x[ch, seg*L + k]
// ---------------------------------------------------------------------------
__global__ void __launch_bounds__(256) seg_carry_kernel(
    const float* __restrict__ x, const float* __restrict__ b0p,
    const float* __restrict__ b1p, const float* __restrict__ a1p,
    float* __restrict__ carry, int Bch, int T, int L)
{
    int tid = blockIdx.x * blockDim.x + threadIdx.x;
    if (tid >= Bch * NSEG) return;
    int ch  = tid & (Bch - 1);     // channel fastest -> coalesced carry write
    int seg = tid / Bch;

    float a1c = clamp1(a1p[0]);
    float c   = b1p[0] + a1c * b0p[0];
    float a2 = a1c * a1c, a3 = a2 * a1c, a4 = a2 * a2;

    const float4* xr = (const float4*)(x + (size_t)ch * T + (size_t)seg * L);
    float s = 0.0f;
    int n4 = L >> 2;
    for (int i = 0; i < n4; ++i) {   // forward Horner, 4 steps fused
        float4 v = xr[i];
        s = a4 * s + c * (a3 * v.x + a2 * v.y + a1c * v.z + v.w);
    }
    carry[(size_t)seg * Bch + ch] = s;
}

// ---------------------------------------------------------------------------
// Pass 2: per channel: scan over NSEG segments. s_in[seg] = entry state.
// Also produces final_state (= d_out tail).
// ---------------------------------------------------------------------------
__global__ void __launch_bounds__(256) seg_scan_kernel(
    const float* __restrict__ state, const float* __restrict__ a1p,
    const float* __restrict__ carry, float* __restrict__ s_in,
    float* __restrict__ final_state, int Bch, int L)
{
    int ch = blockIdx.x * blockDim.x + threadIdx.x;
    if (ch >= Bch) return;
    float a1c = clamp1(a1p[0]);
    float aL = 1.0f;
    for (int i = 0; i < L; ++i) aL *= a1c;   // a1^L (underflows to 0 harmlessly)

    float s = state[ch];
    for (int seg = 0; seg < NSEG; ++seg) {
        s_in[(size_t)seg * Bch + ch] = s;
        s = aL * s + carry[(size_t)seg * Bch + ch];
    }
    final_state[ch] = s;
}

// ---------------------------------------------------------------------------
// Pass 3: one wave per (16-channel strip, segment). Each 16x16 tile:
//   D = M @ X + C   via 4x V_WMMA_F32_16X16X4_F32
// A (16x4 f32) layout: lane<16 -> M=lane, K={4j,4j+1}; lane>=16 -> M=lane-16,
//   K={4j+2,4j+3} (VGPR0=first K of pair, VGPR1=second). [ISA 7.12.2]
// B (4x16 f32): N = lane&15; lanes0-15 hold K={4j,4j+1}, lanes16-31 K={4j+2,4j+3}.
// C/D (16x16 f32): vgpr r, lanes0-15 -> row r; lanes16-31 -> row r+8.
// ---------------------------------------------------------------------------
__global__ void __launch_bounds__(256) wmma_scan_kernel(
    const float* __restrict__ x, const float* __restrict__ b0p,
    const float* __restrict__ b1p, const float* __restrict__ a1p,
    const float* __restrict__ s_in, float* __restrict__ out,
    int Bch, int T, int L)
{
    const int lane = threadIdx.x & 31;
    const int wave = threadIdx.x >> 5;
    const int gw   = blockIdx.x * WAVES_PER_BLOCK + wave;
    const int nstrips = Bch >> 4;
    const int strip = gw & (nstrips - 1);
    const int seg   = gw / nstrips;
    const int h  = (lane >> 4) & 1;     // half-wave selector
    const int r  = lane & 15;           // A row / B,C,D column (channel-in-strip)
    const int ch = strip * 16 + r;

    const float b0  = b0p[0];
    const float b1  = b1p[0];
    const float a1c = clamp1(a1p[0]);
    const float c   = b1 + a1c * b0;

    // powers of a1
    float ap[16];
    ap[0] = 1.0f;
#pragma unroll
    for (int i = 1; i < 16; ++i) ap[i] = ap[i - 1] * a1c;

    // A operands for the 4 K-slices: M[r,k] = (k>r)?0 : (k==r)?b0 : c*a1^(r-1-k)
    v2f A[4];
#pragma unroll
    for (int j = 0; j < 4; ++j) {
        int k0 = 4 * j + 2 * h;
        int k1 = k0 + 1;
        float m0 = (k0 > r) ? 0.0f : ((k0 == r) ? b0 : c * ap[r - 1 - k0]);
        float m1 = (k1 > r) ? 0.0f : ((k1 == r) ? b0 : c * ap[r - 1 - k1]);
        A[j].x = m0;
        A[j].y = m1;
    }

    // carry weights per D vgpr: row m = i + 8h  ->  a1^m
    float pw[8];
#pragma unroll
    for (int i = 0; i < 8; ++i) pw[i] = ap[i + 8 * h];

    float s0 = s_in[(size_t)seg * Bch + ch];   // both half-waves load ch=lane&15

    const float* xrow = x   + (size_t)ch * T + (size_t)seg * L;
    float*       orow = out + (size_t)ch * T + (size_t)seg * L;

    const int ntiles = L >> 4;
    for (int tile = 0; tile < ntiles; ++tile) {
        const int t0 = tile * 16;

        // B operands: lane holds x[k, ch] for k = 4j + 2h + {0,1}
        v2f xb[4];
#pragma unroll
        for (int j = 0; j < 4; ++j)
            xb[j] = *(const v2f*)(xrow + t0 + 4 * j + 2 * h);

        // C accumulator carries the state: C[row, ch] = a1^row * s0
        v8f acc;
#pragma unroll
        for (int i = 0; i < 8; ++i) acc[i] = pw[i] * s0;

        // D = sum_j A_j x B_j + C   (chained f32 WMMA)
#pragma unroll
        for (int j = 0; j < 4; ++j)
            acc = __builtin_amdgcn_wmma_f32_16x16x4_f32(
                false, A[j], false, xb[j], (short)0, acc, false, false);

        // D vgpr i -> out[t0 + i + 8h, ch]: 8 consecutive t -> two B128 stores
        float4 o0 = {acc[0], acc[1], acc[2], acc[3]};
        float4 o1 = {acc[4], acc[5], acc[6], acc[7]};
        float4* op = (float4*)(orow + t0 + 8 * h);
        op[0] = o0;
        op[1] = o1;

        // carry update, exact reference form of the last step:
        //   s_next = b1*x[15] + a1c*out[15]
        // x[15] = xb[3].y and out[15] = acc[7], valid in upper half-wave.
        float v  = b1 * xb[3].y + a1c * acc[7];
        float vl = __shfl_xor(v, 16, 32);   // cross half-wave broadcast
        s0 = h ? v : vl;
    }
}

// ---------------------------------------------------------------------------
// Launch: inputs are (x[B,T], state[B], b0[1], b1[1], a1[1]);
// d_out = outs[B,T] flat then final_state[B].
// ---------------------------------------------------------------------------
extern "C" void kernel_launch(void* const* d_in, const int* in_sizes, int n_in,
                              void* d_out, int out_size, void* d_ws, size_t ws_size,
                              hipStream_t stream) {
    const float* x     = (const float*)d_in[0];
    const float* state = (const float*)d_in[1];
    const float* b0    = (const float*)d_in[2];
    const float* b1    = (const float*)d_in[3];
    const float* a1    = (const float*)d_in[4];
    float* out = (float*)d_out;

    const int Bch = in_sizes[1];            // 4096
    const int T   = in_sizes[0] / Bch;      // 16384
    const int L   = T / NSEG;               // 512 (32 tiles of 16)

    float* carry = (float*)d_ws;                       // [NSEG, B]
    float* s_in  = carry + (size_t)NSEG * Bch;         // [NSEG, B]
    float* final_state = out + (size_t)Bch * T;

    int total1 = Bch * NSEG;
    seg_carry_kernel<<<(total1 + 255) / 256, 256, 0, stream>>>(
        x, b0, b1, a1, carry, Bch, T, L);

    seg_scan_kernel<<<(Bch + 255) / 256, 256, 0, stream>>>(
        state, a1, carry, s_in, final_state, Bch, L);

    int waves  = (Bch / 16) * NSEG;                    // 8192
    int blocks = waves / WAVES_PER_BLOCK;              // 1024
    wmma_scan_kernel<<<blocks, 256, 0, stream>>>(
        x, b0, b1, a1, s_in, out, Bch, T, L);
}